// BarlowTwinsLoss_34892314313040
// MI455X (gfx1250) — compile-verified
//
#include <hip/hip_runtime.h>
#include <hip/hip_bf16.h>

#define LAMBDA_COEFF 0.005f
#define B_SZ 128
#define D_SZ 1024

typedef __attribute__((ext_vector_type(2))) float v2f;
typedef __attribute__((ext_vector_type(8))) float v8f;

// ---------------------------------------------------------------------------
// Phase 1: per-column mean / inv-std (ddof=1) over the batch (B=128 rows),
// computed with V_WMMA_F32_16X16X4_F32 as a ones-vector GEMV reduction.
// One wave (32 threads) handles one 16-column tile of one input matrix.
//   A (16x4)  = all ones            -> D[m,n] = sum_k B[k,n]  (every row m)
//   B (4x16)  = tile of Z (or Z*Z)  -> column partial sums, k += 4 per wmma
// grid.x = 2 * (D/16) = 128 blocks of 32 threads.
// ---------------------------------------------------------------------------
__global__ __launch_bounds__(32) void bt_col_stats_wmma(
    const float* __restrict__ z1, const float* __restrict__ z2,
    float* __restrict__ mean1, float* __restrict__ inv1,
    float* __restrict__ mean2, float* __restrict__ inv2)
{
    const int lane  = threadIdx.x;        // 0..31, one wave per block
    const int which = blockIdx.x >> 6;    // 0 -> z1, 1 -> z2
    const int tile  = blockIdx.x & 63;    // 16-column tile index

    const float* __restrict__ Z = which ? z2 : z1;
    float* __restrict__ meanOut = which ? mean2 : mean1;
    float* __restrict__ invOut  = which ? inv2  : inv1;

    // B-matrix (4x16, KxN) lane layout, symmetric to the documented 16x4 A
    // layout: lanes 0-15 hold N=lane with K = {0,1} in VGPR {0,1};
    // lanes 16-31 hold N=lane-16 with K = {2,3}.
    const int n   = lane & 15;
    const int k0  = (lane >> 4) << 1;     // 0 or 2
    const int col = tile * 16 + n;

    v8f acc_s = {};                       // column sums
    v8f acc_q = {};                       // column sums of squares
    const v2f ones = {1.0f, 1.0f};        // A = ones for every lane/VGPR

    for (int kb = 0; kb < B_SZ; kb += 4) {
        float x0 = Z[(size_t)(kb + k0)     * D_SZ + col];
        float x1 = Z[(size_t)(kb + k0 + 1) * D_SZ + col];
        v2f bv = {x0, x1};
        v2f bq = {x0 * x0, x1 * x1};
        // 8 args: (neg_a, A, neg_b, B, c_mod, C, reuse_a, reuse_b)
        acc_s = __builtin_amdgcn_wmma_f32_16x16x4_f32(
                    false, ones, false, bv, (short)0, acc_s, false, false);
        acc_q = __builtin_amdgcn_wmma_f32_16x16x4_f32(
                    false, ones, false, bq, (short)0, acc_q, false, false);
    }

    // C/D layout: VGPR0 = (M=0, N=lane&15) — all M rows identical since A=1.
    float s = acc_s[0];
    float q = acc_q[0];
    float mean = s * (1.0f / B_SZ);
    float var  = (q - s * s * (1.0f / B_SZ)) * (1.0f / (B_SZ - 1)); // ddof=1
    float inv  = rsqrtf(var);
    if (lane < 16) {
        meanOut[col] = mean;
        invOut[col]  = inv;
    }
}

// ---------------------------------------------------------------------------
// Phase 2: one block per sample b.  Closed-form Barlow-Twins loss from the
// five O(D) row reductions (see analysis).  Wave32 shuffle + LDS reduction.
// ---------------------------------------------------------------------------
__global__ __launch_bounds__(256) void bt_per_sample(
    const float* __restrict__ z1, const float* __restrict__ z2,
    const float* __restrict__ mean1, const float* __restrict__ inv1,
    const float* __restrict__ mean2, const float* __restrict__ inv2,
    float* __restrict__ out)
{
    const int b   = blockIdx.x;
    const int tid = threadIdx.x;

    __shared__ float s_abcb[2];
    __shared__ float s_red[8 * 5];

    const float* __restrict__ r1 = z1 + (size_t)b * D_SZ;
    const float* __restrict__ r2 = z2 + (size_t)b * D_SZ;

    // gfx1250: global_prefetch_b8 hints for the two sample rows (4 KB each).
    __builtin_prefetch(&r1[tid * 4], 0, 1);
    __builtin_prefetch(&r2[tid * 4], 0, 1);

    if (tid == 0) {
        float ab = (r1[b] - mean1[b]) * inv1[b];
        float cb = (r2[b] - mean2[b]) * inv2[b];
        s_abcb[0] = ab;
        s_abcb[1] = cb;
    }
    __syncthreads();
    const float ab = s_abcb[0];
    const float cb = s_abcb[1];

    float T = 0.f, S2 = 0.f, Tc = 0.f, P = 0.f, Q = 0.f;
    #pragma unroll
    for (int jj = 0; jj < D_SZ / 256; ++jj) {
        int j = tid + jj * 256;
        float a = (r1[j] - mean1[j]) * inv1[j];
        float c = (r2[j] - mean2[j]) * inv2[j];
        float a2 = a * a, c2 = c * c;
        T  += a2;
        S2 += c2;
        Tc += a2 * c2;
        float p  = a * c - 1.0f;   P += p * p;
        float qv = ab * c - 1.0f;  float q2 = qv * qv;  Q += q2 * q2;
    }

    // wave32 intra-wave reduction
    #pragma unroll
    for (int off = 16; off > 0; off >>= 1) {
        T  += __shfl_down(T,  off, 32);
        S2 += __shfl_down(S2, off, 32);
        Tc += __shfl_down(Tc, off, 32);
        P  += __shfl_down(P,  off, 32);
        Q  += __shfl_down(Q,  off, 32);
    }
    const int wave = tid >> 5;            // 8 waves per 256-thread block
    if ((tid & 31) == 0) {
        s_red[wave]      = T;
        s_red[8  + wave] = S2;
        s_red[16 + wave] = Tc;
        s_red[24 + wave] = P;
        s_red[32 + wave] = Q;
    }
    __syncthreads();

    if (tid == 0) {
        float t_ = 0.f, s2_ = 0.f, tc_ = 0.f, p_ = 0.f, q_ = 0.f;
        #pragma unroll
        for (int w = 0; w < 8; ++w) {
            t_  += s_red[w];
            s2_ += s_red[8  + w];
            tc_ += s_red[16 + w];
            p_  += s_red[24 + w];
            q_  += s_red[32 + w];
        }
        float t  = ab * cb - 1.0f;
        float d1 = t * t;                          // c[b,b,b] after step 1
        float on_diag  = (p_ - d1) + (d1 - 1.0f) * (d1 - 1.0f);
        float off_main = t_ * s2_ - tc_ - ab * ab * (s2_ - cb * cb);
        float off_spec = q_ - d1 * d1;
        out[b] = on_diag + LAMBDA_COEFF * (off_main + off_spec);
    }
}

extern "C" void kernel_launch(void* const* d_in, const int* in_sizes, int n_in,
                              void* d_out, int out_size, void* d_ws, size_t ws_size,
                              hipStream_t stream) {
    (void)in_sizes; (void)n_in; (void)out_size; (void)ws_size;

    const float* z1 = (const float*)d_in[0];
    const float* z2 = (const float*)d_in[1];

    float* ws    = (float*)d_ws;          // 4 * 1024 floats = 16 KB scratch
    float* mean1 = ws;
    float* inv1  = ws + 1 * D_SZ;
    float* mean2 = ws + 2 * D_SZ;
    float* inv2  = ws + 3 * D_SZ;

    // Phase 1: column stats via WMMA ones-GEMV (64 tiles x 2 matrices).
    bt_col_stats_wmma<<<dim3(128), dim3(32), 0, stream>>>(
        z1, z2, mean1, inv1, mean2, inv2);

    // Phase 2: one block per sample.
    bt_per_sample<<<dim3(B_SZ), dim3(256), 0, stream>>>(
        z1, z2, mean1, inv1, mean2, inv2, (float*)d_out);
}